// SparseRollingFDGRegressor_19842748907726
// MI455X (gfx1250) — compile-verified
//
#include <hip/hip_runtime.h>
#include <hip/hip_bf16.h>
#include <math.h>

#define NN 2000
#define BB 4
#define DIN 64
#define LL 256
#define RNK 32
#define DH 256
#define KSEL 20
#define EPSV 1e-6f
#define CTILE 5   // column tiles per wave in gram kernel: 16x80 strip, 2000 = 25*80

typedef __attribute__((ext_vector_type(2))) float v2f;
typedef __attribute__((ext_vector_type(8))) float v8f;

// ---------------- prep: bottleneck MLP + S*Bmat, R projections -------------
__global__ __launch_bounds__(64)
void prep_kernel(const float* __restrict__ X,
                 const float* __restrict__ We1, const float* __restrict__ be1,
                 const float* __restrict__ We2, const float* __restrict__ be2,
                 const float* __restrict__ Ws,  const float* __restrict__ Wr,
                 const float* __restrict__ Bm,
                 float* __restrict__ Xm, float* __restrict__ SB, float* __restrict__ R) {
  __shared__ float xs[DIN], hs[DIN], xm[DIN], ss[RNK];
  const int row = blockIdx.x;     // 0 .. B*N-1
  const int t = threadIdx.x;      // 0 .. 63
  xs[t] = X[(size_t)row * DIN + t];
  __syncthreads();
  float acc = be1[t];
  for (int d = 0; d < DIN; ++d) acc += xs[d] * We1[d * DIN + t];
  hs[t] = fmaxf(acc, 0.f);
  __syncthreads();
  acc = be2[t];
  for (int f = 0; f < DIN; ++f) acc += hs[f] * We2[f * DIN + t];
  xm[t] = acc;
  Xm[(size_t)row * DIN + t] = acc;
  __syncthreads();
  if (t < RNK) {
    float s = 0.f, r = 0.f;
    for (int d = 0; d < DIN; ++d) { s += xm[d] * Ws[d * RNK + t]; r += xm[d] * Wr[d * RNK + t]; }
    ss[t] = s;
    R[(size_t)row * RNK + t] = r;
  }
  __syncthreads();
  if (t < RNK) {
    float sb = 0.f;
    for (int k = 0; k < RNK; ++k) sb += ss[k] * Bm[k * RNK + t];
    SB[(size_t)row * RNK + t] = sb;
  }
}

// ---------------- history normalization: hn = centered/rms / sqrt(L) -------
__global__ __launch_bounds__(256)
void hist_kernel(const float* __restrict__ hist, float* __restrict__ hn) {
  __shared__ float red[LL];
  const int row = blockIdx.x;
  const int t = threadIdx.x;
  const float v = hist[(size_t)row * LL + t];
  red[t] = v; __syncthreads();
  for (int s = LL / 2; s > 0; s >>= 1) { if (t < s) red[t] += red[t + s]; __syncthreads(); }
  const float mean = red[0] * (1.0f / LL);
  __syncthreads();
  const float c = v - mean;
  red[t] = c * c; __syncthreads();
  for (int s = LL / 2; s > 0; s >>= 1) { if (t < s) red[t] += red[t + s]; __syncthreads(); }
  float denom = fmaxf(sqrtf(red[0] * (1.0f / LL)), EPSV);
  // sim = (norm . norm)/L  ->  pre-scale each row by 1/sqrt(L)=1/16
  hn[(size_t)row * LL + t] = (c / denom) * (1.0f / 16.0f);
}

// ---------------- dense gram via V_WMMA_F32_16X16X4_F32 --------------------
// Each wave produces a 16x(16*CTILE) strip: A fragment loaded once per k-step,
// reused across CTILE accumulators (1.2 loads/WMMA instead of 2.0).
// mode 0: sigmoid(dot) (tau=1);  mode 1: relu(dot).  Diagonal zeroed.
__global__ __launch_bounds__(32)
void gram_kernel(const float* __restrict__ Arows, const float* __restrict__ Brows,
                 float* __restrict__ dense, int K, int mode) {
  const int tm = blockIdx.x;           // row tile (16 rows)
  const int tn = blockIdx.y;           // col strip (16*CTILE cols)
  const int lane = threadIdx.x;        // 0..31
  const int half = lane >> 4;          // 0 | 1
  const int l16 = lane & 15;
  const float* ap = Arows + (size_t)(tm * 16 + l16) * K;     // A row (M = l%16)
  const float* bp[CTILE];
  for (int s = 0; s < CTILE; ++s)      // B=R^T col (N = l%16) per sub-tile
    bp[s] = Brows + (size_t)(tn * 16 * CTILE + s * 16 + l16) * K;
  v8f c[CTILE];
  for (int s = 0; s < CTILE; ++s) c[s] = (v8f){};
  for (int k = 0; k < K; k += 4) {
    // A 16x4 f32: lanes 0-15 hold K = k,k+1 ; lanes 16-31 hold K = k+2,k+3
    v2f a = *(const v2f*)(ap + k + 2 * half);
    for (int s = 0; s < CTILE; ++s) {
      v2f b = *(const v2f*)(bp[s] + k + 2 * half);
      c[s] = __builtin_amdgcn_wmma_f32_16x16x4_f32(false, a, false, b, (short)0, c[s], false, false);
    }
  }
  for (int s = 0; s < CTILE; ++s) {
    const int col = tn * 16 * CTILE + s * 16 + l16;
    for (int v = 0; v < 8; ++v) {
      const int row = tm * 16 + v + 8 * half;   // D: VGPR v -> M = v + 8*(lane/16)
      float val = c[s][v];
      if (mode == 0) val = 1.0f / (1.0f + __expf(-val));
      else           val = fmaxf(val, 0.0f);
      if (row == col) val = 0.0f;
      dense[(size_t)row * NN + col] = val;
    }
  }
}

// ---------------- top-20 per row + row-normalize ---------------------------
__global__ __launch_bounds__(256)
void topk_kernel(const float* __restrict__ dense,
                 float* __restrict__ ov, int* __restrict__ oi, int row_base) {
  __shared__ float buf[NN];
  __shared__ float sv[256];
  __shared__ int   si[256];
  __shared__ float kv[KSEL];
  __shared__ int   ki[KSEL];
  const int r = blockIdx.x;
  const int t = threadIdx.x;
  for (int i = t; i < NN; i += 256) buf[i] = dense[(size_t)r * NN + i];
  __syncthreads();
  for (int k = 0; k < KSEL; ++k) {
    float best = -INFINITY; int bi = NN;
    for (int i = t; i < NN; i += 256) {
      const float v = buf[i];
      if (v > best) { best = v; bi = i; }   // first (lowest) index wins ties
    }
    sv[t] = best; si[t] = bi; __syncthreads();
    for (int s = 128; s > 0; s >>= 1) {
      if (t < s) {
        if (sv[t + s] > sv[t] || (sv[t + s] == sv[t] && si[t + s] < si[t])) {
          sv[t] = sv[t + s]; si[t] = si[t + s];
        }
      }
      __syncthreads();
    }
    if (t == 0) { kv[k] = sv[0]; ki[k] = si[0]; if (si[0] < NN) buf[si[0]] = -INFINITY; }
    __syncthreads();
  }
  if (t == 0) {
    float s = 0.f;
    for (int k = 0; k < KSEL; ++k) s += kv[k];
    sv[0] = fmaxf(s, EPSV);
  }
  __syncthreads();
  if (t < KSEL) {
    ov[(size_t)(row_base + r) * KSEL + t] = kv[t] / sv[0];
    oi[(size_t)(row_base + r) * KSEL + t] = ki[t];
  }
}

// ---------------- mix branches into 40-nnz rows, renormalize ---------------
__global__ void merge_kernel(const float* __restrict__ fv, const int* __restrict__ fi,
                             const float* __restrict__ rv, const int* __restrict__ ri,
                             const float* __restrict__ mix_logit,
                             float* __restrict__ av, int* __restrict__ ai, int total) {
  const int row = blockIdx.x * blockDim.x + threadIdx.x;
  if (row >= total) return;
  const float mix = 1.0f / (1.0f + __expf(-mix_logit[0]));
  float vals[2 * KSEL]; int idxs[2 * KSEL];
  float s = 0.f;
  for (int j = 0; j < KSEL; ++j) {
    const float v = mix * fv[(size_t)row * KSEL + j];
    vals[j] = v; idxs[j] = fi[(size_t)row * KSEL + j]; s += v;
  }
  for (int j = 0; j < KSEL; ++j) {
    const float v = (1.0f - mix) * rv[(size_t)row * KSEL + j];
    vals[KSEL + j] = v; idxs[KSEL + j] = ri[(size_t)row * KSEL + j]; s += v;
  }
  s = fmaxf(s, EPSV);
  for (int j = 0; j < 2 * KSEL; ++j) {
    av[(size_t)row * 2 * KSEL + j] = vals[j] / s;
    ai[(size_t)row * 2 * KSEL + j] = idxs[j];
  }
}

// ---------------- sparse A @ X_model ---------------------------------------
__global__ __launch_bounds__(64)
void ax_kernel(const float* __restrict__ av, const int* __restrict__ ai,
               const float* __restrict__ Xm, float* __restrict__ AX) {
  const int row = blockIdx.x;       // global row b*N+n
  const int b = row / NN;
  const int d = threadIdx.x;
  float acc = 0.f;
  for (int j = 0; j < 2 * KSEL; ++j) {
    const float a = av[(size_t)row * 2 * KSEL + j];
    const int m = ai[(size_t)row * 2 * KSEL + j];
    acc += a * Xm[((size_t)(b * NN + m)) * DIN + d];
  }
  AX[(size_t)row * DIN + d] = acc;
}

// ---------------- GNN hidden + W2 dot --------------------------------------
__global__ __launch_bounds__(256)
void hidden_kernel(const float* __restrict__ AX, const float* __restrict__ W1,
                   const float* __restrict__ b1, const float* __restrict__ W2,
                   float* __restrict__ z) {
  __shared__ float xs[DIN];
  __shared__ float red[DH];
  const int row = blockIdx.x;
  const int t = threadIdx.x;
  if (t < DIN) xs[t] = AX[(size_t)row * DIN + t];
  __syncthreads();
  float acc = b1[t];
  for (int d = 0; d < DIN; ++d) acc += xs[d] * W1[d * DH + t];
  red[t] = fmaxf(acc, 0.f) * W2[t];
  __syncthreads();
  for (int s = DH / 2; s > 0; s >>= 1) { if (t < s) red[t] += red[t + s]; __syncthreads(); }
  if (t == 0) z[row] = red[0];
}

// ---------------- final sparse A @ z + b2 ----------------------------------
__global__ void final_kernel(const float* __restrict__ av, const int* __restrict__ ai,
                             const float* __restrict__ z, const float* __restrict__ b2,
                             float* __restrict__ out, int total) {
  const int row = blockIdx.x * blockDim.x + threadIdx.x;
  if (row >= total) return;
  const int b = row / NN;
  float acc = b2[0];
  for (int j = 0; j < 2 * KSEL; ++j)
    acc += av[(size_t)row * 2 * KSEL + j] * z[b * NN + ai[(size_t)row * 2 * KSEL + j]];
  out[row] = acc;
}

extern "C" void kernel_launch(void* const* d_in, const int* in_sizes, int n_in,
                              void* d_out, int out_size, void* d_ws, size_t ws_size,
                              hipStream_t stream) {
  (void)in_sizes; (void)n_in; (void)out_size; (void)ws_size;
  const float* X       = (const float*)d_in[0];
  const float* history = (const float*)d_in[1];
  const float* We1     = (const float*)d_in[2];
  const float* be1     = (const float*)d_in[3];
  const float* We2     = (const float*)d_in[4];
  const float* be2     = (const float*)d_in[5];
  const float* Ws      = (const float*)d_in[6];
  const float* Wr      = (const float*)d_in[7];
  const float* Bm      = (const float*)d_in[8];
  const float* mixl    = (const float*)d_in[9];
  const float* W1      = (const float*)d_in[10];
  const float* b1      = (const float*)d_in[11];
  const float* W2      = (const float*)d_in[12];
  const float* b2      = (const float*)d_in[13];
  float* out = (float*)d_out;

  float* ws = (float*)d_ws;
  size_t off = 0;
  float* Xm    = ws + off; off += (size_t)BB * NN * DIN;   // X_model
  float* SBm   = ws + off; off += (size_t)BB * NN * RNK;   // S @ Bmat
  float* Rm    = ws + off; off += (size_t)BB * NN * RNK;   // R
  float* hn    = ws + off; off += (size_t)BB * NN * LL;    // normalized history
  float* dense = ws + off; off += (size_t)NN * NN;         // per-batch dense scores
  float* fv    = ws + off; off += (size_t)BB * NN * KSEL;
  int*   fi    = (int*)(ws + off); off += (size_t)BB * NN * KSEL;
  float* rv    = ws + off; off += (size_t)BB * NN * KSEL;
  int*   ri    = (int*)(ws + off); off += (size_t)BB * NN * KSEL;
  float* av    = ws + off; off += (size_t)BB * NN * 2 * KSEL;
  int*   ai    = (int*)(ws + off); off += (size_t)BB * NN * 2 * KSEL;
  float* AX    = ws + off; off += (size_t)BB * NN * DIN;
  float* zz    = ws + off; off += (size_t)BB * NN;

  const int total = BB * NN;
  prep_kernel<<<total, 64, 0, stream>>>(X, We1, be1, We2, be2, Ws, Wr, Bm, Xm, SBm, Rm);
  hist_kernel<<<total, 256, 0, stream>>>(history, hn);

  dim3 gtile(NN / 16, NN / (16 * CTILE));   // 125 x 25 waves
  for (int b = 0; b < BB; ++b) {
    // FDG branch: sigmoid(SB . R^T), zero diag, top-20
    gram_kernel<<<gtile, 32, 0, stream>>>(SBm + (size_t)b * NN * RNK,
                                          Rm  + (size_t)b * NN * RNK, dense, RNK, 0);
    topk_kernel<<<NN, 256, 0, stream>>>(dense, fv, fi, b * NN);
    // Rolling-correlation branch: relu(hn . hn^T), zero diag, top-20
    gram_kernel<<<gtile, 32, 0, stream>>>(hn + (size_t)b * NN * LL,
                                          hn + (size_t)b * NN * LL, dense, LL, 1);
    topk_kernel<<<NN, 256, 0, stream>>>(dense, rv, ri, b * NN);
  }

  merge_kernel<<<(total + 255) / 256, 256, 0, stream>>>(fv, fi, rv, ri, mixl, av, ai, total);
  ax_kernel<<<total, 64, 0, stream>>>(av, ai, Xm, AX);
  hidden_kernel<<<total, 256, 0, stream>>>(AX, W1, b1, W2, zz);
  final_kernel<<<(total + 255) / 256, 256, 0, stream>>>(av, ai, zz, b2, out, total);
}